// DynamicMaskHead_77730318123068
// MI455X (gfx1250) — compile-verified
//
#include <hip/hip_runtime.h>
#include <math.h>

#define DEVINL __device__ __forceinline__

typedef __attribute__((ext_vector_type(2))) float v2f;
typedef __attribute__((ext_vector_type(8))) float v8f;

constexpr int H = 100, W = 152, HW = H * W;
constexpr int CIN = 8;
constexpr int NGEN = 169;          // per-head generated params
constexpr int PSTR = 2 * NGEN + 81; // 419 params per instance
constexpr int OH = 200, OW = 304;  // factor-2 aligned-bilinear output
constexpr int TILES = HW / 16;     // 950 (exact)

// ---------------------------------------------------------------- helpers

DEVINL float swz16(float x) {
  // group-of-32 swizzle: src_lane = lane ^ 16  (xor=0x10, or=0, and=0x1f)
  return __int_as_float(__builtin_amdgcn_ds_swizzle(__float_as_int(x), 0x401f));
}

DEVINL v2f mkv2(float a, float b) { v2f r; r.x = a; r.y = b; return r; }

DEVINL v8f wmma4(v2f a, v2f b, v8f c) {
  // D = A(16x4,f32) * B(4x16,f32) + C(16x16,f32)
  return __builtin_amdgcn_wmma_f32_16x16x4_f32(
      /*neg_a=*/false, a, /*neg_b=*/false, b,
      /*c_mod=*/(short)0, c, /*reuse_a=*/false, /*reuse_b=*/false);
}

// A-matrix element generators (block-diagonal body/edge packing).
// Layer1: A1[m][k], m: 0-7 body rows, 8-15 edge rows; K layout (24):
//   0..1 rel, 2..9 fb, 10..11 pad, 12..13 rel, 14..21 fe, 22..23 pad
DEVINL float a1elem(const float* p, int m, int k) {
  if (m < 8) return (k < 10) ? p[m * 10 + k] : 0.f;
  int kk = k - 12;
  return (kk >= 0 && kk < 10) ? p[NGEN + (m - 8) * 10 + kk] : 0.f;
}
// Layer2: diag(W2_b, W2_e), K=16
DEVINL float a2elem(const float* p, int m, int k) {
  if (m < 8) return (k < 8) ? p[80 + m * 8 + k] : 0.f;
  return (k >= 8) ? p[NGEN + 80 + (m - 8) * 8 + (k - 8)] : 0.f;
}
// Fused layer1: Wf1 in rows 0-7, K=8
DEVINL float afelem(const float* p, int m, int k) {
  return (m < 8) ? p[338 + m * 8 + k] : 0.f;
}

// ---------------------------------------------------------------- kernel 1
// Per 16-pixel tile: 12x v_wmma_f32_16x16x4_f32 computes all three heads'
// logits for 16 pixels; writes fp32 logits to ws[3][n_inst][HW].

__global__ void __launch_bounds__(128)
mask_head_wmma_kernel(const float* __restrict__ fb, const float* __restrict__ fe,
                      const float* __restrict__ params,
                      const float* __restrict__ loc, const float* __restrict__ soi,
                      const int* __restrict__ im_inds, const int* __restrict__ lvls,
                      const int* __restrict__ stride_p,
                      float* __restrict__ ws, int n_inst) {
  const int inst = blockIdx.x;
  const int lane = threadIdx.x & 31;
  const int wid  = blockIdx.y * (blockDim.x >> 5) + (threadIdx.x >> 5);
  const int nw   = gridDim.y * (blockDim.x >> 5);
  const int m    = lane & 15;
  const bool hi  = lane >= 16;

  const float* p  = params + (size_t)inst * PSTR;
  const float* pe = p + NGEN;

  // --- Build per-wave constant A matrices (16x4 f32 layout:
  //     lane l: M = l%16; v.x -> K = 4c + (l<16?0:2), v.y -> +1)
  v2f A1[6], A2[4], AF[2];
#pragma unroll
  for (int c = 0; c < 6; ++c) {
    int k0 = 4 * c + (hi ? 2 : 0);
    A1[c] = mkv2(a1elem(p, m, k0), a1elem(p, m, k0 + 1));
  }
#pragma unroll
  for (int c = 0; c < 4; ++c) {
    int k0 = 4 * c + (hi ? 2 : 0);
    A2[c] = mkv2(a2elem(p, m, k0), a2elem(p, m, k0 + 1));
  }
#pragma unroll
  for (int c = 0; c < 2; ++c) {
    int k0 = 4 * c + (hi ? 2 : 0);
    AF[c] = mkv2(afelem(p, m, k0), afelem(p, m, k0 + 1));
  }

  // --- Bias accumulators in C/D layout (lane<16: rows v (body);
  //     lane>=16: rows v+8 (edge))
  v8f C1, C2, CF;
#pragma unroll
  for (int v = 0; v < 8; ++v) {
    C1[v] = hi ? pe[152 + v] : p[152 + v];
    C2[v] = hi ? pe[160 + v] : p[160 + v];
    CF[v] = hi ? 0.f : p[410 + v];
  }
  float W3[8], WF2[8];
#pragma unroll
  for (int k = 0; k < 8; ++k) {
    W3[k]  = hi ? pe[144 + k] : p[144 + k];  // body (lo lanes) / edge (hi lanes)
    WF2[k] = p[402 + k];
  }
  const float B3  = hi ? pe[168] : p[168];
  const float BF2 = p[418];

  const int img = im_inds[inst];
  const float* fbI = fb + (size_t)img * CIN * HW;
  const float* feI = fe + (size_t)img * CIN * HW;
  const float locx = loc[2 * inst], locy = loc[2 * inst + 1];
  const float inv_s = 1.f / soi[lvls[inst]];
  const int stride = stride_p[0];
  const int shalf = stride >> 1;

  float* wsb = ws + (size_t)inst * HW;
  float* wse = wsb + (size_t)n_inst * HW;
  float* wsf = wse + (size_t)n_inst * HW;

  // channel bases for the B-matrix halves (lo lanes feed K=4c..+1, hi lanes K=4c+2..+3)
  const int cb = hi ? 0 : 2;
  const float* fbc = fbI + (size_t)cb * HW;
  const float* fec = feI + (size_t)cb * HW;

  for (int t = wid; t < TILES; t += nw) {
    const int pix = t * 16 + m;           // HW == 950*16 exactly, no tail
    const int py = pix / W;
    const int px = pix - py * W;
    const float relx = (locx - (float)(px * stride + shalf)) * inv_s;
    const float rely = (locy - (float)(py * stride + shalf)) * inv_s;

    const float b0 = fbc[pix];
    const float b1 = fbc[HW + pix];
    const float b4 = fbc[4 * HW + pix];
    const float b5 = fbc[5 * HW + pix];
    const float e0 = fec[pix];
    const float e1 = fec[HW + pix];
    const float e4 = fec[4 * HW + pix];
    const float e5 = fec[5 * HW + pix];

    // ---- layer 1: D = [a1_b ; a1_e] (pre-relu), K=24 over 6 chunks
    v8f d = C1;
    d = wmma4(A1[0], mkv2(hi ? b0 : relx, hi ? b1 : rely), d);
    d = wmma4(A1[1], mkv2(hi ? b4 : b0,   hi ? b5 : b1),   d);
    d = wmma4(A1[2], mkv2(hi ? 0.f : b4,  hi ? 0.f : b5),  d);
    d = wmma4(A1[3], mkv2(hi ? e0 : relx, hi ? e1 : rely), d);
    d = wmma4(A1[4], mkv2(hi ? e4 : e0,   hi ? e5 : e1),   d);
    d = wmma4(A1[5], mkv2(hi ? 0.f : e4,  hi ? 0.f : e5),  d);
#pragma unroll
    for (int v = 0; v < 8; ++v) d[v] = fmaxf(d[v], 0.f);

    // C-layout -> B-layout: need the other lane-half's rows
    const float o0 = swz16(d[0]), o1 = swz16(d[1]), o2 = swz16(d[2]), o3 = swz16(d[3]);
    const float o4 = swz16(d[4]), o5 = swz16(d[5]), o6 = swz16(d[6]), o7 = swz16(d[7]);

    // ---- layer 2: D2 = [a2_b ; a2_e], K=16 over 4 chunks
    v8f d2 = C2;
    d2 = wmma4(A2[0], mkv2(hi ? o2 : d[0],   hi ? o3 : d[1]),   d2);
    d2 = wmma4(A2[1], mkv2(hi ? o6 : d[4],   hi ? o7 : d[5]),   d2);
    d2 = wmma4(A2[2], mkv2(hi ? d[2] : o0,   hi ? d[3] : o1),   d2);
    d2 = wmma4(A2[3], mkv2(hi ? d[6] : o4,   hi ? d[7] : o5),   d2);
#pragma unroll
    for (int v = 0; v < 8; ++v) d2[v] = fmaxf(d2[v], 0.f);

    // body (lanes 0-15) / edge (lanes 16-31) logits: 1x8 dot
    float acc = B3;
#pragma unroll
    for (int k = 0; k < 8; ++k) acc = fmaf(W3[k], d2[k], acc);

    // s = a2_b + a2_e: both lane halves end up holding s columns
    float sv[8];
#pragma unroll
    for (int v = 0; v < 8; ++v) sv[v] = d2[v] + swz16(d2[v]);

    // ---- fused layer 1: K=8 over 2 chunks
    v8f df = CF;
    df = wmma4(AF[0], mkv2(hi ? sv[2] : sv[0], hi ? sv[3] : sv[1]), df);
    df = wmma4(AF[1], mkv2(hi ? sv[6] : sv[4], hi ? sv[7] : sv[5]), df);

    float accf = BF2;
#pragma unroll
    for (int k = 0; k < 8; ++k) accf = fmaf(WF2[k], fmaxf(df[k], 0.f), accf);

    if (!hi) {
      wsb[pix] = acc;   // logit_body
      wsf[pix] = accf;  // logit_fused
    } else {
      wse[pix] = acc;   // logit_edge
    }
  }
}

// ---------------------------------------------------------------- kernel 2
// aligned_bilinear(factor=2) + sigmoid; pure bandwidth (scratch is L2-hot).
// out[y,x] = interp at src_y = max(y-1,0)/2, src_x = max(x-1,0)/2, edge-clamped.

__global__ void __launch_bounds__(256)
upsample_sigmoid_kernel(const float* __restrict__ ws, float* __restrict__ out,
                        int n_inst) {
  const long long total = 3LL * n_inst * OH * OW;
  long long idx = (long long)blockIdx.x * blockDim.x + threadIdx.x;
  if (idx >= total) return;
  int x = (int)(idx % OW);
  int y = (int)((idx / OW) % OH);
  long long rest = idx / ((long long)OW * OH);
  int i = (int)(rest % n_inst);
  int t = (int)(rest / n_inst);

  const int dy = y > 0 ? y - 1 : 0;
  const int dx = x > 0 ? x - 1 : 0;
  const int y0 = dy >> 1, x0 = dx >> 1;
  const float fy = (dy & 1) ? 0.5f : 0.f;
  const float fx = (dx & 1) ? 0.5f : 0.f;
  const int y1 = (y0 + 1 < H) ? y0 + 1 : H - 1;
  const int x1 = (x0 + 1 < W) ? x0 + 1 : W - 1;

  const float* b = ws + ((size_t)t * n_inst + i) * HW;
  const float v00 = b[y0 * W + x0];
  const float v01 = b[y0 * W + x1];
  const float v10 = b[y1 * W + x0];
  const float v11 = b[y1 * W + x1];
  const float top = v00 + fx * (v01 - v00);
  const float bot = v10 + fx * (v11 - v10);
  const float v = top + fy * (bot - top);
  out[idx] = 1.f / (1.f + expf(-v));
}

// ---------------------------------------------------------------- fallback
// Scalar path used only if d_ws is too small: per output pixel, recompute
// the <=4 source-pixel MLPs directly.

DEVINL void mlp3_eval(const float* __restrict__ p,
                      const float* __restrict__ fbI, const float* __restrict__ feI,
                      float locx, float locy, float inv_s, int stride, int shalf,
                      int yy, int xx, float o[3]) {
  const int pix = yy * W + xx;
  float xb[10], xe[10];
  xb[0] = xe[0] = (locx - (float)(xx * stride + shalf)) * inv_s;
  xb[1] = xe[1] = (locy - (float)(yy * stride + shalf)) * inv_s;
#pragma unroll
  for (int c = 0; c < 8; ++c) {
    xb[2 + c] = fbI[c * HW + pix];
    xe[2 + c] = feI[c * HW + pix];
  }
  float ab[8], ae[8];
#pragma unroll
  for (int q = 0; q < 8; ++q) {
    float sb = p[152 + q], se = p[NGEN + 152 + q];
#pragma unroll
    for (int k = 0; k < 10; ++k) {
      sb = fmaf(p[q * 10 + k], xb[k], sb);
      se = fmaf(p[NGEN + q * 10 + k], xe[k], se);
    }
    ab[q] = fmaxf(sb, 0.f);
    ae[q] = fmaxf(se, 0.f);
  }
  float a2b[8], a2e[8];
#pragma unroll
  for (int q = 0; q < 8; ++q) {
    float sb = p[160 + q], se = p[NGEN + 160 + q];
#pragma unroll
    for (int k = 0; k < 8; ++k) {
      sb = fmaf(p[80 + q * 8 + k], ab[k], sb);
      se = fmaf(p[NGEN + 80 + q * 8 + k], ae[k], se);
    }
    a2b[q] = fmaxf(sb, 0.f);
    a2e[q] = fmaxf(se, 0.f);
  }
  float lb = p[168], le = p[NGEN + 168];
#pragma unroll
  for (int k = 0; k < 8; ++k) {
    lb = fmaf(p[144 + k], a2b[k], lb);
    le = fmaf(p[NGEN + 144 + k], a2e[k], le);
  }
  float lf = p[418];
#pragma unroll
  for (int q = 0; q < 8; ++q) {
    float sf = p[410 + q];
#pragma unroll
    for (int k = 0; k < 8; ++k)
      sf = fmaf(p[338 + q * 8 + k], a2b[k] + a2e[k], sf);
    lf = fmaf(p[402 + q], fmaxf(sf, 0.f), lf);
  }
  o[0] = lb; o[1] = le; o[2] = lf;
}

__global__ void __launch_bounds__(128)
fallback_kernel(const float* __restrict__ fb, const float* __restrict__ fe,
                const float* __restrict__ params,
                const float* __restrict__ loc, const float* __restrict__ soi,
                const int* __restrict__ im_inds, const int* __restrict__ lvls,
                const int* __restrict__ stride_p,
                float* __restrict__ out, int n_inst) {
  const long long total = (long long)n_inst * OH * OW;
  long long idx = (long long)blockIdx.x * blockDim.x + threadIdx.x;
  if (idx >= total) return;
  const int x = (int)(idx % OW);
  const int y = (int)((idx / OW) % OH);
  const int i = (int)(idx / ((long long)OW * OH));

  const float* p = params + (size_t)i * PSTR;
  const int img = im_inds[i];
  const float* fbI = fb + (size_t)img * CIN * HW;
  const float* feI = fe + (size_t)img * CIN * HW;
  const float locx = loc[2 * i], locy = loc[2 * i + 1];
  const float inv_s = 1.f / soi[lvls[i]];
  const int stride = stride_p[0];
  const int shalf = stride >> 1;

  const int dy = y > 0 ? y - 1 : 0;
  const int dx = x > 0 ? x - 1 : 0;
  const int y0 = dy >> 1, x0 = dx >> 1;
  const float fy = (dy & 1) ? 0.5f : 0.f;
  const float fx = (dx & 1) ? 0.5f : 0.f;
  const int y1 = (y0 + 1 < H) ? y0 + 1 : H - 1;
  const int x1 = (x0 + 1 < W) ? x0 + 1 : W - 1;

  const float wgt[4] = {(1.f - fy) * (1.f - fx), (1.f - fy) * fx,
                        fy * (1.f - fx), fy * fx};
  const int ys[4] = {y0, y0, y1, y1};
  const int xs[4] = {x0, x1, x0, x1};

  float acc[3] = {0.f, 0.f, 0.f};
  for (int tap = 0; tap < 4; ++tap) {
    if (wgt[tap] > 0.f) {
      float o[3];
      mlp3_eval(p, fbI, feI, locx, locy, inv_s, stride, shalf, ys[tap], xs[tap], o);
      acc[0] = fmaf(wgt[tap], o[0], acc[0]);
      acc[1] = fmaf(wgt[tap], o[1], acc[1]);
      acc[2] = fmaf(wgt[tap], o[2], acc[2]);
    }
  }
#pragma unroll
  for (int t = 0; t < 3; ++t)
    out[(size_t)t * total + idx] = 1.f / (1.f + expf(-acc[t]));
}

// ---------------------------------------------------------------- launch

extern "C" void kernel_launch(void* const* d_in, const int* in_sizes, int n_in,
                              void* d_out, int out_size, void* d_ws, size_t ws_size,
                              hipStream_t stream) {
  (void)n_in; (void)out_size;
  const float* fb     = (const float*)d_in[0];
  const float* fe     = (const float*)d_in[1];
  const float* params = (const float*)d_in[2];
  const float* loc    = (const float*)d_in[3];
  const float* soi    = (const float*)d_in[4];
  const int*   imi    = (const int*)d_in[5];
  const int*   lvl    = (const int*)d_in[6];
  const int*   strp   = (const int*)d_in[7];
  const int n_inst = in_sizes[2] / PSTR;
  float* out = (float*)d_out;

  const size_t need = (size_t)3 * n_inst * HW * sizeof(float);
  if (ws_size >= need) {
    dim3 grid1(n_inst, 8, 1);
    mask_head_wmma_kernel<<<grid1, 128, 0, stream>>>(
        fb, fe, params, loc, soi, imi, lvl, strp, (float*)d_ws, n_inst);
    const long long total = 3LL * n_inst * OH * OW;
    const int blocks = (int)((total + 255) / 256);
    upsample_sigmoid_kernel<<<blocks, 256, 0, stream>>>(
        (const float*)d_ws, out, n_inst);
  } else {
    const long long total = (long long)n_inst * OH * OW;
    const int blocks = (int)((total + 127) / 128);
    fallback_kernel<<<blocks, 128, 0, stream>>>(
        fb, fe, params, loc, soi, imi, lvl, strp, out, n_inst);
  }
}